// Channelatt_18408229831231
// MI455X (gfx1250) — compile-verified
//
#include <hip/hip_runtime.h>
#include <math.h>

#define NB 32
#define CC 256
#define HH 56
#define WW 56
#define HWS (HH * WW)      // 3136
#define LN_EPS 1e-5f

typedef __attribute__((ext_vector_type(2))) float v2f;
typedef __attribute__((ext_vector_type(8))) float v8f;

// ---------------------------------------------------------------------------
// Pass 1: logits[n][s] = sum_c x[n,c,s] * gc_w[c] + gc_b
// Consecutive threads -> consecutive s -> fully coalesced; gc_w[c] is a
// wave-uniform scalar load. This is the first (HBM) pass over x; afterwards
// x is L2-resident (103MB < 192MB L2).
// ---------------------------------------------------------------------------
__global__ void k_logits(const float* __restrict__ x,
                         const float* __restrict__ gc_w,
                         const float* __restrict__ gc_b,
                         float* __restrict__ logits) {
    int s = blockIdx.x * blockDim.x + threadIdx.x;
    int n = blockIdx.y;
    if (s >= HWS) return;
    const float* xp = x + (size_t)n * CC * HWS + s;
    float acc = 0.f;
#pragma unroll 4
    for (int c = 0; c < CC; ++c) acc = fmaf(xp[(size_t)c * HWS], gc_w[c], acc);
    logits[n * HWS + s] = acc + gc_b[0];
}

// ---------------------------------------------------------------------------
// Pass 2a: softmax over space per image, normalized in place. Tiny (N*HW).
// ---------------------------------------------------------------------------
__global__ void k_softmax(float* __restrict__ logits) {
    __shared__ float red[256];
    int n = blockIdx.x, t = threadIdx.x;
    float* row = logits + n * HWS;

    float mx = -INFINITY;
    for (int s = t; s < HWS; s += 256) mx = fmaxf(mx, row[s]);
    red[t] = mx; __syncthreads();
    for (int o = 128; o > 0; o >>= 1) { if (t < o) red[t] = fmaxf(red[t], red[t + o]); __syncthreads(); }
    mx = red[0]; __syncthreads();

    float sum = 0.f;
    for (int s = t; s < HWS; s += 256) sum += __expf(row[s] - mx);
    red[t] = sum; __syncthreads();
    for (int o = 128; o > 0; o >>= 1) { if (t < o) red[t] += red[t + o]; __syncthreads(); }
    float inv = 1.f / red[0];

    for (int s = t; s < HWS; s += 256) row[s] = __expf(row[s] - mx) * inv;
}

// ---------------------------------------------------------------------------
// Pass 2b: fused pooling per (n,c): weighted-sum (x_g), max (x_max),
// multi-spectral DCT (x_dct). DCT basis built in LDS per block; group
// i = c/64 -> (u,v) = [(0,0),(0,1),(1,0),(1,1)][i]. Reads x from L2.
// ---------------------------------------------------------------------------
__global__ void k_pool(const float* __restrict__ x,
                       const float* __restrict__ attn,
                       float* __restrict__ xg,
                       float* __restrict__ xmax,
                       float* __restrict__ xdct) {
    __shared__ float bh[HH];
    __shared__ float bw[WW];
    __shared__ float red[256];
    int c = blockIdx.x, n = blockIdx.y, t = threadIdx.x;
    int grp = c >> 6;
    int u = grp >> 1, v = grp & 1;
    if (t < HH) {
        float su = (u != 0) ? 1.41421356237f : 1.f;
        float sv = (v != 0) ? 1.41421356237f : 1.f;
        bh[t] = cosf((float)M_PI * (float)u * (t + 0.5f) / (float)HH) * (su / sqrtf((float)HH));
        bw[t] = cosf((float)M_PI * (float)v * (t + 0.5f) / (float)WW) * (sv / sqrtf((float)WW));
    }
    __syncthreads();

    const float* xp = x + ((size_t)n * CC + c) * HWS;
    const float* ap = attn + n * HWS;
    float g = 0.f, mx = -INFINITY, d = 0.f;
    for (int s = t; s < HWS; s += 256) {
        float val = xp[s];
        g  = fmaf(val, ap[s], g);
        mx = fmaxf(mx, val);
        d  = fmaf(val, bh[s / WW] * bw[s % WW], d);
    }
    red[t] = g; __syncthreads();
    for (int o = 128; o > 0; o >>= 1) { if (t < o) red[t] += red[t + o]; __syncthreads(); }
    if (t == 0) xg[n * CC + c] = red[0];
    __syncthreads();
    red[t] = mx; __syncthreads();
    for (int o = 128; o > 0; o >>= 1) { if (t < o) red[t] = fmaxf(red[t], red[t + o]); __syncthreads(); }
    if (t == 0) xmax[n * CC + c] = red[0];
    __syncthreads();
    red[t] = d; __syncthreads();
    for (int o = 128; o > 0; o >>= 1) { if (t < o) red[t] += red[t + o]; __syncthreads(); }
    if (t == 0) xdct[n * CC + c] = red[0];
}

// ---------------------------------------------------------------------------
// Blend + channel conv1d(k=3, zero pad) + LayerNorm over C. One block per n,
// one thread per channel.
// ---------------------------------------------------------------------------
__global__ void k_mix_ln(const float* __restrict__ xg,
                         const float* __restrict__ xmax,
                         const float* __restrict__ xdct,
                         const float* __restrict__ wmax,
                         const float* __restrict__ wdct,
                         const float* __restrict__ lc_w,
                         const float* __restrict__ lc_b,
                         const float* __restrict__ lcln_g,
                         const float* __restrict__ lcln_b,
                         float* __restrict__ att_ln) {
    __shared__ float xs[CC];
    __shared__ float red[CC];
    int n = blockIdx.x, c = threadIdx.x;
    float s = wmax[c] * xmax[n * CC + c] + wdct[c] * xdct[n * CC + c];
    xs[c] = s; __syncthreads();
    float left  = (c > 0)      ? xs[c - 1] : 0.f;
    float right = (c < CC - 1) ? xs[c + 1] : 0.f;
    float xl = lc_w[0] * left + lc_w[1] * s + lc_w[2] * right + lc_b[0];
    float t = xg[n * CC + c] + s + xl;

    red[c] = t; __syncthreads();
    for (int o = 128; o > 0; o >>= 1) { if (c < o) red[c] += red[c + o]; __syncthreads(); }
    float mu = red[0] * (1.f / CC); __syncthreads();
    float dv = t - mu;
    red[c] = dv * dv; __syncthreads();
    for (int o = 128; o > 0; o >>= 1) { if (c < o) red[c] += red[c + o]; __syncthreads(); }
    float var = red[0] * (1.f / CC);
    att_ln[n * CC + c] = dv * rsqrtf(var + LN_EPS) * lcln_g[c] + lcln_b[c];
}

// ---------------------------------------------------------------------------
// WMMA f32 16x16x4: att2 = att_ln @ tw_w.T + tw_b, then LN over C + sigmoid.
// 32 output tiles (2 M-tiles x 16 N-tiles) = 32 waves = one 1024-thread WG.
// A fragments from LDS (att_ln, 32KB), B fragments from tw_w (L2-resident).
// VGPR layouts per CDNA5 ISA 7.12.2 for 32-bit A(16x4)/B(4x16)/C(16x16).
// LN done per row by the matching wave using wave32 shuffle reductions.
// ---------------------------------------------------------------------------
__global__ __launch_bounds__(1024) void k_gate_wmma(const float* __restrict__ att_ln,
                                                    const float* __restrict__ tw_w,
                                                    const float* __restrict__ tw_b,
                                                    const float* __restrict__ twln_g,
                                                    const float* __restrict__ twln_b,
                                                    float* __restrict__ gate) {
    __shared__ float sA[NB * CC];   // 32 KB
    int tid = threadIdx.x;
    for (int i = tid; i < NB * CC; i += 1024) sA[i] = att_ln[i];
    __syncthreads();

    int wave = tid >> 5;
    int lane = tid & 31;
    int mt = wave >> 4;          // 0..1  (M tile: rows n)
    int nt = wave & 15;          // 0..15 (N tile: output channels j)
    int lrow  = lane & 15;
    int khalf = (lane >> 4) << 1;   // lanes 0-15 -> K+0/K+1, lanes 16-31 -> K+2/K+3

    const float* Arow = sA + (mt * 16 + lrow) * CC;               // A[M=lrow][k]
    const float* Brow = tw_w + (size_t)(nt * 16 + lrow) * CC;     // B[k][N=lrow] = tw_w[j][k]

    v8f acc = {};
    for (int k = 0; k < CC; k += 4) {
        int kb = k + khalf;
        v2f a; a.x = Arow[kb]; a.y = Arow[kb + 1];
        v2f b; b.x = Brow[kb]; b.y = Brow[kb + 1];
        acc = __builtin_amdgcn_wmma_f32_16x16x4_f32(
            /*neg_a=*/false, a, /*neg_b=*/false, b,
            /*c_mod=*/(short)0, acc, /*reuse_a=*/false, /*reuse_b=*/false);
    }
    __syncthreads();   // everyone done reading sA before we overwrite it

    // Scatter D tile (+bias) back to LDS: VGPR r -> M=r (lanes 0-15) / M=r+8
    int rowoff = (lane >> 4) * 8;
#pragma unroll
    for (int r = 0; r < 8; ++r) {
        int row = mt * 16 + rowoff + r;
        int j = nt * 16 + lrow;
        sA[row * CC + j] = acc[r] + tw_b[j];
    }
    __syncthreads();

    // LayerNorm over C + sigmoid: wave w owns row n=w (32 waves == 32 rows).
    float sum = 0.f, sq = 0.f;
    float vals[8];
#pragma unroll
    for (int i = 0; i < 8; ++i) {
        float vv = sA[wave * CC + lane + i * 32];
        vals[i] = vv; sum += vv; sq += vv * vv;
    }
#pragma unroll
    for (int m = 16; m > 0; m >>= 1) {
        sum += __shfl_xor(sum, m, 32);
        sq  += __shfl_xor(sq,  m, 32);
    }
    float mu  = sum * (1.f / CC);
    float var = sq * (1.f / CC) - mu * mu;
    float rs  = rsqrtf(var + LN_EPS);
#pragma unroll
    for (int i = 0; i < 8; ++i) {
        int j = lane + i * 32;
        float g = (vals[i] - mu) * rs * twln_g[j] + twln_b[j];
        gate[wave * CC + j] = 1.f / (1.f + __expf(-g));
    }
}

// ---------------------------------------------------------------------------
// Pass 3: out = x * gate[n,c]; float4-vectorized (HW divisible by 4, so the
// channel index is uniform within each float4).
// ---------------------------------------------------------------------------
__global__ void k_scale(const float4* __restrict__ x4,
                        const float* __restrict__ gate,
                        float4* __restrict__ out4, int nvec) {
    int i = blockIdx.x * blockDim.x + threadIdx.x;
    if (i >= nvec) return;
    int nc = (i * 4) / HWS;          // flat (n*C + c)
    float g = gate[nc];
    float4 v = x4[i];
    v.x *= g; v.y *= g; v.z *= g; v.w *= g;
    out4[i] = v;
}

extern "C" void kernel_launch(void* const* d_in, const int* in_sizes, int n_in,
                              void* d_out, int out_size, void* d_ws, size_t ws_size,
                              hipStream_t stream) {
    const float* x      = (const float*)d_in[0];
    const float* gc_w   = (const float*)d_in[1];
    const float* gc_b   = (const float*)d_in[2];
    const float* lc_w   = (const float*)d_in[3];
    const float* lc_b   = (const float*)d_in[4];
    const float* lcln_g = (const float*)d_in[5];
    const float* lcln_b = (const float*)d_in[6];
    const float* tw_w   = (const float*)d_in[7];
    const float* tw_b   = (const float*)d_in[8];
    const float* twln_g = (const float*)d_in[9];
    const float* twln_b = (const float*)d_in[10];
    const float* wdct   = (const float*)d_in[11];
    const float* wmax   = (const float*)d_in[12];

    float* ws     = (float*)d_ws;
    float* logits = ws;                   // N*HW (reused in-place as attn)
    float* xg     = logits + NB * HWS;    // N*C
    float* xmax   = xg     + NB * CC;     // N*C
    float* xdct   = xmax   + NB * CC;     // N*C
    float* attln  = xdct   + NB * CC;     // N*C
    float* gate   = attln  + NB * CC;     // N*C

    dim3 g1((HWS + 255) / 256, NB);
    k_logits<<<g1, 256, 0, stream>>>(x, gc_w, gc_b, logits);

    k_softmax<<<NB, 256, 0, stream>>>(logits);

    dim3 g3(CC, NB);
    k_pool<<<g3, 256, 0, stream>>>(x, logits, xg, xmax, xdct);

    k_mix_ln<<<NB, CC, 0, stream>>>(xg, xmax, xdct, wmax, wdct,
                                    lc_w, lc_b, lcln_g, lcln_b, attln);

    k_gate_wmma<<<1, 1024, 0, stream>>>(attln, tw_w, tw_b, twln_g, twln_b, gate);

    int nvec = NB * CC * HWS / 4;
    k_scale<<<(nvec + 255) / 256, 256, 0, stream>>>((const float4*)x, gate,
                                                    (float4*)d_out, nvec);
}